// acoustic_Shrinkage_new_43722767073471
// MI455X (gfx1250) — compile-verified
//
#include <hip/hip_runtime.h>
#include <cmath>

#define B_ 32
#define C_ 256
#define L_ 8192
#define BN_EPS 1e-5f

typedef float v2f __attribute__((ext_vector_type(2)));
typedef float v8f __attribute__((ext_vector_type(8)));

__device__ __forceinline__ float db_xform(float x) {
    return 20.0f * log10f(x + 1.0f);
}

// ---------------------------------------------------------------------------
// Kernel 1: per (b,c) row of L=8192: g = mean|y|, min, max, exact median via
// 4-pass radix select on float bit patterns staged in LDS (all |y| >= 0).
// ---------------------------------------------------------------------------
__global__ __launch_bounds__(256)
void k_rowstats(const float* __restrict__ x,
                float* __restrict__ g, float* __restrict__ med,
                float* __restrict__ mn, float* __restrict__ mx)
{
    __shared__ __align__(16) unsigned sdata[L_];   // 32 KB: |y| bit patterns
    __shared__ float sred[256];
    __shared__ unsigned shist[256];
    __shared__ unsigned ssel[2];

    const int row = blockIdx.x;          // 0..8191
    const int t   = threadIdx.x;         // 0..255
    const float4* xr = (const float4*)(x + (size_t)row * L_);

    float sum = 0.f, vmin = 3.0e38f, vmax = -3.0e38f;
    #pragma unroll
    for (int i = 0; i < 8; ++i) {
        const int f4 = t + 256 * i;
        float4 v = xr[f4];
        float y0 = fabsf(db_xform(v.x));
        float y1 = fabsf(db_xform(v.y));
        float y2 = fabsf(db_xform(v.z));
        float y3 = fabsf(db_xform(v.w));
        sum += (y0 + y1) + (y2 + y3);
        vmin = fminf(vmin, fminf(fminf(y0, y1), fminf(y2, y3)));
        vmax = fmaxf(vmax, fmaxf(fmaxf(y0, y1), fmaxf(y2, y3)));
        uint4 u;
        u.x = __float_as_uint(y0); u.y = __float_as_uint(y1);
        u.z = __float_as_uint(y2); u.w = __float_as_uint(y3);
        ((uint4*)sdata)[f4] = u;
    }

    // --- block reductions: sum, min, max ---
    sred[t] = sum; __syncthreads();
    for (int ofs = 128; ofs > 0; ofs >>= 1) {
        if (t < ofs) sred[t] += sred[t + ofs];
        __syncthreads();
    }
    if (t == 0) g[row] = sred[0] * (1.0f / (float)L_);
    __syncthreads();

    sred[t] = vmin; __syncthreads();
    for (int ofs = 128; ofs > 0; ofs >>= 1) {
        if (t < ofs) sred[t] = fminf(sred[t], sred[t + ofs]);
        __syncthreads();
    }
    if (t == 0) mn[row] = sred[0];
    __syncthreads();

    sred[t] = vmax; __syncthreads();
    for (int ofs = 128; ofs > 0; ofs >>= 1) {
        if (t < ofs) sred[t] = fmaxf(sred[t], sred[t + ofs]);
        __syncthreads();
    }
    if (t == 0) mx[row] = sred[0];
    __syncthreads();

    // --- radix select: k-th smallest (k = 4096, 1-indexed) == s[4095] ---
    unsigned prefix = 0;
    int k = L_ / 2;                                  // 4096
    const int shifts[4] = {23, 15, 7, 0};
    #pragma unroll
    for (int p = 0; p < 4; ++p) {
        const int shift = shifts[p];
        const int width = (p == 3) ? 7 : 8;
        const int nb = 1 << width;
        const unsigned himask = 0xFFFFFFFFu << (shift + width);

        shist[t] = 0u;
        __syncthreads();
        for (int i = 0; i < 32; ++i) {
            unsigned e = sdata[t + 256 * i];
            if ((e & himask) == prefix)
                atomicAdd(&shist[(e >> shift) & (unsigned)(nb - 1)], 1u);
        }
        __syncthreads();
        // inclusive scan over nb bins (Hillis-Steele)
        for (int ofs = 1; ofs < nb; ofs <<= 1) {
            unsigned add = (t < nb && t >= ofs) ? shist[t - ofs] : 0u;
            __syncthreads();
            if (t < nb) shist[t] += add;
            __syncthreads();
        }
        if (t < nb) {
            unsigned c  = shist[t];
            unsigned cp = (t == 0) ? 0u : shist[t - 1];
            if (c >= (unsigned)k && cp < (unsigned)k) { ssel[0] = (unsigned)t; ssel[1] = cp; }
        }
        __syncthreads();
        prefix |= (ssel[0] << shift);
        k -= (int)ssel[1];
        __syncthreads();
    }
    if (t == 0) med[row] = __uint_as_float(prefix);
}

// ---------------------------------------------------------------------------
// Kernel 2: single workgroup (32 waves), full fp32 WMMA for both GEMMs.
//   h = g @ W1^T + b1 ; BatchNorm over batch ; ReLU ;
//   alpha = sigmoid(h @ W2^T + b2) ; T = threshold select.
// Wave w owns output tile (m0, n0) = (16*(w/16), 16*(w%16)).
// A frag (16x4 f32): lane holds M = lane%16, VGPR v holds K = 2*(lane/16)+v.
// B frag (4x16 f32): lane holds N = lane%16, same K split (mirrors A).
// C/D frag: VGPR v -> M = 8*(lane/16)+v, N = lane%16.
// ---------------------------------------------------------------------------
__global__ __launch_bounds__(1024)
void k_mid(const float* __restrict__ g,
           const float* __restrict__ W1, const float* __restrict__ b1,
           const float* __restrict__ gamma, const float* __restrict__ beta,
           const float* __restrict__ W2, const float* __restrict__ b2,
           const float* __restrict__ med, const float* __restrict__ mn,
           const float* __restrict__ mx, float* __restrict__ Tout)
{
    __shared__ float sh[B_ * C_];     // 32 KB: h then normalized h
    __shared__ float smu[C_], sscale[C_];

    const int tid = threadIdx.x;
    const int w   = tid >> 5;         // wave 0..31
    const int t   = tid & 31;         // lane
    const int m0  = (w >> 4) << 4;    // 0 or 16
    const int n0  = (w & 15) << 4;    // 0..240
    const int lm  = t & 15;
    const int q   = (t >> 4) << 1;    // K sub-offset: 0 or 2
    const int s8  = (t >> 4) << 3;    // C/D M sub-offset: 0 or 8
    const int mrow = m0 + lm;         // A row (batch)
    const int ncol = n0 + lm;         // B/D column (channel)

    // ---- GEMM1: h = g @ W1^T + b1 ----
    v8f acc;
    {
        const float bias = b1[ncol];
        #pragma unroll
        for (int v = 0; v < 8; ++v) acc[v] = bias;
    }
    for (int k = 0; k < C_; k += 4) {
        v2f a = *(const v2f*)(g  + mrow * C_ + k + q);
        v2f b = *(const v2f*)(W1 + ncol * C_ + k + q);
        acc = __builtin_amdgcn_wmma_f32_16x16x4_f32(
                  false, a, false, b, (short)0, acc, false, false);
    }
    #pragma unroll
    for (int v = 0; v < 8; ++v)
        sh[(m0 + s8 + v) * C_ + n0 + lm] = acc[v];
    __syncthreads();

    // ---- BatchNorm over batch (32) per channel ----
    if (tid < C_) {
        const int j = tid;
        float s = 0.f;
        for (int b = 0; b < B_; ++b) s += sh[b * C_ + j];
        const float mu = s * (1.0f / (float)B_);
        float s2 = 0.f;
        for (int b = 0; b < B_; ++b) { float d = sh[b * C_ + j] - mu; s2 += d * d; }
        const float var = s2 * (1.0f / (float)B_);
        smu[j]    = mu;
        sscale[j] = gamma[j] / sqrtf(var + BN_EPS);
    }
    __syncthreads();
    // normalize + ReLU in place (each element touched by exactly one thread)
    for (int i = tid; i < B_ * C_; i += 1024) {
        const int j = i & (C_ - 1);
        float hv = (sh[i] - smu[j]) * sscale[j] + beta[j];
        sh[i] = fmaxf(hv, 0.f);
    }
    __syncthreads();

    // ---- GEMM2: pre = hn @ W2^T + b2 ----
    v8f acc2;
    {
        const float bias = b2[ncol];
        #pragma unroll
        for (int v = 0; v < 8; ++v) acc2[v] = bias;
    }
    for (int k = 0; k < C_; k += 4) {
        v2f a = *(const v2f*)(&sh[mrow * C_ + k + q]);
        v2f b = *(const v2f*)(W2 + ncol * C_ + k + q);
        acc2 = __builtin_amdgcn_wmma_f32_16x16x4_f32(
                   false, a, false, b, (short)0, acc2, false, false);
    }

    // ---- alpha = sigmoid(pre); T = threshold select ----
    #pragma unroll
    for (int v = 0; v < 8; ++v) {
        const int bb = m0 + s8 + v;            // batch
        const int r  = bb * C_ + n0 + lm;      // (b, c) flat
        const float al = 1.0f / (1.0f + expf(-acc2[v]));
        const float m_ = med[r], lo = mn[r], hi = mx[r];
        // where(mx-med > med-mn, med + a*(mn-med), med + a*(mx-med))
        const float T = (hi - m_ > m_ - lo) ? (m_ + al * (lo - m_))
                                            : (m_ + al * (hi - m_));
        Tout[r] = T;
    }
}

// ---------------------------------------------------------------------------
// Kernel 3: out = sign(y) * max(|y| - T[row], 0), float4 vectorized.
// 8 blocks per row (8192 floats / (256 thr * 4)); T is block-uniform (SGPR).
// ---------------------------------------------------------------------------
__device__ __forceinline__ float shrink1(float xc, float Tv) {
    float y = db_xform(xc);
    float a = fmaxf(fabsf(y) - Tv, 0.f);
    return (y > 0.f) ? a : ((y < 0.f) ? -a : 0.f);
}

__global__ __launch_bounds__(256)
void k_apply(const float* __restrict__ x, const float* __restrict__ T,
             float* __restrict__ out)
{
    const int f = blockIdx.x * 256 + threadIdx.x;  // float4 index
    const float Tv = T[blockIdx.x >> 3];           // uniform per block
    float4 xv = ((const float4*)x)[f];
    float4 o;
    o.x = shrink1(xv.x, Tv);
    o.y = shrink1(xv.y, Tv);
    o.z = shrink1(xv.z, Tv);
    o.w = shrink1(xv.w, Tv);
    ((float4*)out)[f] = o;
}

// ---------------------------------------------------------------------------
extern "C" void kernel_launch(void* const* d_in, const int* in_sizes, int n_in,
                              void* d_out, int out_size, void* d_ws, size_t ws_size,
                              hipStream_t stream)
{
    const float* x     = (const float*)d_in[0];
    const float* W1    = (const float*)d_in[1];
    const float* b1    = (const float*)d_in[2];
    const float* gamma = (const float*)d_in[3];
    const float* beta  = (const float*)d_in[4];
    const float* W2    = (const float*)d_in[5];
    const float* b2    = (const float*)d_in[6];
    float* outp = (float*)d_out;

    float* w   = (float*)d_ws;     // 5 * 8192 floats = 160 KB scratch
    float* g   = w;
    float* med = w + 8192;
    float* mn  = w + 16384;
    float* mx  = w + 24576;
    float* T   = w + 32768;

    k_rowstats<<<B_ * C_, 256, 0, stream>>>(x, g, med, mn, mx);
    k_mid<<<1, 1024, 0, stream>>>(g, W1, b1, gamma, beta, W2, b2, med, mn, mx, T);
    k_apply<<<(B_ * C_ * L_) / (256 * 4), 256, 0, stream>>>(x, T, outp);
}